// GCNGuard_32950989094961
// MI455X (gfx1250) — compile-verified
//
#include <hip/hip_runtime.h>
#include <math.h>

typedef __attribute__((ext_vector_type(2))) float v2f;
typedef __attribute__((ext_vector_type(8))) float v8f;

// ---------------------------------------------------------------- utilities
__device__ __forceinline__ float wave_reduce_sum(float v) {
  #pragma unroll
  for (int m = 16; m >= 1; m >>= 1) v += __shfl_xor(v, m, 32);
  return v;
}

// ---------------------------------------------------------------- CSR build
// Edges are lexicographically sorted by (row,col) (np.unique semantics), so
// boundary marking gives deterministic row_start[N+1].
__global__ void k_row_start(const int* __restrict__ row, int* __restrict__ row_start,
                            int E, int N) {
  int e = blockIdx.x * blockDim.x + threadIdx.x;
  if (e >= E) return;
  int r  = row[e];
  int rp = (e == 0) ? -1 : row[e - 1];
  for (int n = rp + 1; n <= r; ++n) row_start[n] = e;
  if (e == E - 1) {
    for (int n = r + 1; n <= N; ++n) row_start[n] = E;
  }
}

// ---------------------------------------------------------------- row normalize (D=128)
__global__ void k_normalize(const float* __restrict__ x, float* __restrict__ xn, int N) {
  int n    = (blockIdx.x * blockDim.x + threadIdx.x) >> 5;
  int lane = threadIdx.x & 31;
  if (n >= N) return;
  const float4* xr = (const float4*)(x + (size_t)n * 128);
  float4 v = xr[lane];
  float s  = v.x * v.x + v.y * v.y + v.z * v.z + v.w * v.w;
  s = wave_reduce_sum(s);
  float nrm = sqrtf(s);
  float inv = (nrm == 0.f) ? 1.f : (1.f / nrm);
  float4 o = make_float4(v.x * inv, v.y * inv, v.z * inv, v.w * inv);
  ((float4*)(xn + (size_t)n * 128))[lane] = o;
}

// ---------------------------------------------------------------- per-edge cosine sim (wave/edge)
__global__ void k_edge_sim(const float* __restrict__ xn, const int* __restrict__ row,
                           const int* __restrict__ col, float* __restrict__ sim, int E) {
  int e    = (blockIdx.x * blockDim.x + threadIdx.x) >> 5;
  int lane = threadIdx.x & 31;
  if (e >= E) return;
  int r = row[e], c = col[e];
  float4 a = ((const float4*)(xn + (size_t)r * 128))[lane];
  float4 b = ((const float4*)(xn + (size_t)c * 128))[lane];
  float d = a.x * b.x + a.y * b.y + a.z * b.z + a.w * b.w;
  d = wave_reduce_sum(d);
  if (lane == 0) sim[e] = (d >= 0.1f) ? d : 0.f;   // sim[sim < 0.1] = 0
}

// ---------------------------------------------------------------- per-node L1 sum + degree (wave/node)
__global__ void k_row_reduce(const float* __restrict__ sim, const int* __restrict__ row_start,
                             float* __restrict__ rowsum, float* __restrict__ selfw, int N) {
  int n    = (blockIdx.x * blockDim.x + threadIdx.x) >> 5;
  int lane = threadIdx.x & 31;
  if (n >= N) return;
  int s0 = row_start[n], s1 = row_start[n + 1];
  float sum = 0.f, cnt = 0.f;
  for (int e = s0 + lane; e < s1; e += 32) {
    float s = sim[e];
    sum += s;
    cnt += (s > 0.f) ? 1.f : 0.f;
  }
  sum = wave_reduce_sum(sum);
  cnt = wave_reduce_sum(cnt);
  if (lane == 0) {
    rowsum[n] = sum;
    selfw[n]  = expf(1.f / (cnt + 1.f));   // exp(lam), lam = 1/(deg+1)
  }
}

// ---------------------------------------------------------------- edge weight finalize (in place)
__global__ void k_edge_finalize(float* __restrict__ sim, const float* __restrict__ rowsum,
                                const int* __restrict__ row, int E) {
  int e = blockIdx.x * blockDim.x + threadIdx.x;
  if (e >= E) return;
  float s = sim[e];
  if (s > 0.f) {
    float rs = rowsum[row[e]];           // rs >= s > 0 here
    sim[e] = expf(s / rs);
  } else {
    sim[e] = 0.f;
  }
}

// ---------------------------------------------------------------- WMMA f32 GEMM: out = X(NxK) @ W(KxM) + bias
// One wave computes one 16x16 output tile using V_WMMA_F32_16X16X4_F32.
__global__ void k_gemm_wmma(const float* __restrict__ X, const float* __restrict__ W,
                            const float* __restrict__ bias, float* __restrict__ out,
                            int Nrows, int K, int M) {
  int wid  = (blockIdx.x * blockDim.x + threadIdx.x) >> 5;
  int lane = threadIdx.x & 31;
  int ntc  = M >> 4;                       // column tiles
  int ntr  = (Nrows + 15) >> 4;            // row tiles
  if (wid >= ntc * ntr) return;
  int tn = wid % ntc;
  int tm = wid / ntc;

  int m    = lane & 15;                    // M index within tile (A & C), N index (B & C)
  int half = lane >> 4;
  int kb   = half * 2;                     // K sub-offset for A/B halves

  int arow = tm * 16 + m;
  if (arow >= Nrows) arow = Nrows - 1;     // safe clamp (N is a multiple of 16 here)
  const float* xrow = X + (size_t)arow * K;
  int coln = tn * 16 + m;

  v8f c = {};
  for (int k0 = 0; k0 < K; k0 += 4) {
    v2f a, b;
    a.x = xrow[k0 + kb];
    a.y = xrow[k0 + kb + 1];
    b.x = W[(size_t)(k0 + kb) * M + coln];
    b.y = W[(size_t)(k0 + kb + 1) * M + coln];
    c = __builtin_amdgcn_wmma_f32_16x16x4_f32(false, a, false, b, (short)0, c, false, false);
  }

  float bv = bias[coln];
  #pragma unroll
  for (int v = 0; v < 8; ++v) {
    int rowm = tm * 16 + v + half * 8;     // C layout: VGPR v -> M=v (lanes 0-15) / v+8 (16-31)
    if (rowm < Nrows) out[(size_t)rowm * M + coln] = c[v] + bv;
  }
}

// ---------------------------------------------------------------- aggregation (wave/node, atomic-free)
// out[n] = act( selfw[n]*lin[n] + sum_e w[e]*lin[col[e]] )
template <int PER, bool RELU>
__global__ void k_aggregate(const float* __restrict__ lin, const float* __restrict__ ew,
                            const float* __restrict__ selfw, const int* __restrict__ col,
                            const int* __restrict__ row_start, float* __restrict__ out,
                            int N, int M) {
  int n    = (blockIdx.x * blockDim.x + threadIdx.x) >> 5;
  int lane = threadIdx.x & 31;
  if (n >= N) return;
  int base = lane * PER;

  float acc[PER];
  float sw = selfw[n];
  const float* lr = lin + (size_t)n * M;
  #pragma unroll
  for (int j = 0; j < PER; ++j) acc[j] = sw * lr[base + j];

  int s0 = row_start[n], s1 = row_start[n + 1];
  for (int e = s0; e < s1; ++e) {
    float w = ew[e];
    if (w != 0.f) {
      const float* hr = lin + (size_t)col[e] * M;
      #pragma unroll
      for (int j = 0; j < PER; ++j) acc[j] += w * hr[base + j];
    }
  }

  float* orow = out + (size_t)n * M;
  #pragma unroll
  for (int j = 0; j < PER; ++j) orow[base + j] = RELU ? fmaxf(acc[j], 0.f) : acc[j];
}

// ---------------------------------------------------------------- launcher
extern "C" void kernel_launch(void* const* d_in, const int* in_sizes, int n_in,
                              void* d_out, int out_size, void* d_ws, size_t ws_size,
                              hipStream_t stream) {
  (void)n_in; (void)out_size; (void)ws_size;

  const float* x   = (const float*)d_in[0];
  const int*   ei  = (const int*)d_in[1];
  const float* W1  = (const float*)d_in[2];
  const float* b1  = (const float*)d_in[3];
  const float* W2  = (const float*)d_in[4];
  const float* b2  = (const float*)d_in[5];

  const int D = 128, H = 128, C = 64;
  const int N = in_sizes[0] / D;
  const int E = in_sizes[1] / 2;
  const int* row = ei;
  const int* col = ei + E;

  // workspace layout (floats)
  float* ws     = (float*)d_ws;
  float* xn     = ws;                         // N*128 normalized features
  float* lin    = xn   + (size_t)N * 128;     // N*128 linear output (reused, layer2 uses N*64)
  float* feat2  = lin  + (size_t)N * 128;     // N*128 layer-1 output
  float* sim    = feat2 + (size_t)N * 128;    // E (sim, then edge weights in place)
  float* rowsum = sim  + (size_t)E;           // N
  float* selfw  = rowsum + (size_t)N;         // N
  int*   rstart = (int*)(selfw + (size_t)N);  // N+1

  const int TPB = 128;                        // 4 waves per block
  dim3 blkNode((N + 3) / 4), blkEdgeW((E + 3) / 4);
  dim3 blkEdgeT((E + 255) / 256);

  // CSR row offsets (edges sorted by row -> deterministic)
  k_row_start<<<blkEdgeT, 256, 0, stream>>>(row, rstart, E, N);

  int tiles1 = ((N + 15) / 16) * (H / 16);
  int tiles2 = ((N + 15) / 16) * (C / 16);

  // ---------------- layer 1 ----------------
  k_normalize<<<blkNode, TPB, 0, stream>>>(x, xn, N);
  k_edge_sim<<<blkEdgeW, TPB, 0, stream>>>(xn, row, col, sim, E);
  k_row_reduce<<<blkNode, TPB, 0, stream>>>(sim, rstart, rowsum, selfw, N);
  k_edge_finalize<<<blkEdgeT, 256, 0, stream>>>(sim, rowsum, row, E);
  k_gemm_wmma<<<(tiles1 + 3) / 4, TPB, 0, stream>>>(x, W1, b1, lin, N, D, H);
  k_aggregate<4, true><<<blkNode, TPB, 0, stream>>>(lin, sim, selfw, col, rstart, feat2, N, H);

  // ---------------- layer 2 ----------------
  k_normalize<<<blkNode, TPB, 0, stream>>>(feat2, xn, N);
  k_edge_sim<<<blkEdgeW, TPB, 0, stream>>>(xn, row, col, sim, E);
  k_row_reduce<<<blkNode, TPB, 0, stream>>>(sim, rstart, rowsum, selfw, N);
  k_edge_finalize<<<blkEdgeT, 256, 0, stream>>>(sim, rowsum, row, E);
  k_gemm_wmma<<<(tiles2 + 3) / 4, TPB, 0, stream>>>(feat2, W2, b2, lin, N, H, C);
  k_aggregate<2, false><<<blkNode, TPB, 0, stream>>>(lin, sim, selfw, col, rstart,
                                                     (float*)d_out, N, C);
}